// Group_maskpoint_only_neig_51247549775875
// MI455X (gfx1250) — compile-verified
//
#include <hip/hip_runtime.h>

typedef __attribute__((ext_vector_type(2))) float v2f;
typedef __attribute__((ext_vector_type(8))) float v8f;

#define B_   16
#define N_   8192
#define G_   512
#define M_   32
#define WAVES 4

__global__ __launch_bounds__(128)
void knn_group_wmma_kernel(const float* __restrict__ xyz,
                           const float* __restrict__ center,
                           float* __restrict__ out) {
    const int tid   = threadIdx.x;
    const int lane  = tid & 31;
    const int w     = tid >> 5;          // wave id in block, 0..3
    const int topHalf = lane >> 4;       // 0: lanes 0-15, 1: lanes 16-31
    const int lm    = lane & 15;
    const int b     = blockIdx.x >> 5;   // batch
    const int gbase = (blockIdx.x & 31) * 16;

    // Per-wave top-32 candidate lists for the 16 shared centers.
    __shared__ float lds_d[WAVES][16][32];
    __shared__ int   lds_i[WAVES][16][32];

    const float INFV = __builtin_inff();
    for (int i = tid; i < WAVES * 16 * 32; i += 128) {
        (&lds_d[0][0][0])[i] = INFV;
        (&lds_i[0][0][0])[i] = 0x7FFFFFFF;
    }
    __syncthreads();

    // ---- A matrix (16 centers x K=4), ISA layout for 32-bit A 16x4:
    // lanes 0-15: VGPR0=K0, VGPR1=K1 ; lanes 16-31: VGPR0=K2, VGPR1=K3
    const float* cp = center + ((size_t)b * G_ + gbase + lm) * 3;
    v2f a;
    a.x = (topHalf == 0) ? cp[0] : cp[2];
    a.y = (topHalf == 0) ? cp[1] : 1.0f;

    // Running max (threshold) of each half's center list, per v in 0..7.
    float cm[8]; int am[8];
#pragma unroll
    for (int v = 0; v < 8; ++v) { cm[v] = INFV; am[v] = 0; }

    const int nTiles = (N_ / WAVES) / 16;  // 128 tiles of 16 points per wave
    const int base = w * (N_ / WAVES);     // 2048 points per wave
    const float* pbase = xyz + ((size_t)b * N_ + base + lm) * 3;
#pragma unroll 2
    for (int t = 0; t < nTiles; ++t) {
        const int nb = base + t * 16;
        const float* pp = pbase + (size_t)t * 48;   // 16 points * 3 floats
        if (t + 1 < nTiles) {
            // WGP-scope prefetch (scope 0: pull into all cache levels on miss).
            // Non-speculative at WGP scope, hence the in-bounds guard above.
            const float* nxt = pp + 48;
            asm volatile("global_prefetch_b8 %0, off" :: "v"(nxt));
        }
        float px = pp[0], py = pp[1], pz = pp[2];
        // B matrix (K=4 x 16 points): rows = (-2x, -2y, -2z, |p|^2), branchless
        float n2 = px * px + py * py + pz * pz;
        v2f bv;
        bv.x = -2.0f * ((topHalf == 0) ? px : pz);
        bv.y = (topHalf == 0) ? (-2.0f * py) : n2;

        v8f cz = {};
        // proxy_dist[m][n] = -2 c_m . p_n + |p_n|^2   (order-preserving per center)
        v8f dd = __builtin_amdgcn_wmma_f32_16x16x4_f32(
            false, a, false, bv, (short)0, cz, false, false);

        // D layout: VGPR v, lanes 0-15 -> center (gbase+v), lanes 16-31 -> center (gbase+v+8),
        // point n = nb + (lane&15).
#pragma unroll
        for (int v = 0; v < 8; ++v) {
            const int myc = v + 8 * topHalf;
            float dv = dd[v];
            unsigned mask = (unsigned)__ballot(dv < cm[v]);
            while (mask) {                       // wave-uniform loop
                unsigned lo = mask & 0xFFFFu;
                unsigned hi = mask >> 16;
                int jlo = lo ? (__ffs(lo) - 1) : 0;
                int jhi = hi ? (__ffs(hi) + 15) : 16;
                bool act = topHalf ? (hi != 0) : (lo != 0);
                int  src = topHalf ? jhi : jlo;
                float cdist = __shfl(dv, src, 32);
                int   cidx  = nb + (src & 15);
                if (act && cdist < cm[v]) {
                    if (lm == 0) {               // one writer per half
                        lds_d[w][myc][am[v]] = cdist;
                        lds_i[w][myc][am[v]] = cidx;
                    }
                    // recompute (max, argmax) of the 32-entry list with 16 lanes
                    float e0 = lds_d[w][myc][2 * lm];
                    float e1 = lds_d[w][myc][2 * lm + 1];
                    float md; int ma;
                    if (e1 > e0) { md = e1; ma = 2 * lm + 1; }
                    else         { md = e0; ma = 2 * lm; }
#pragma unroll
                    for (int off = 1; off < 16; off <<= 1) {
                        float od = __shfl_xor(md, off, 32);
                        int   oa = __shfl_xor(ma, off, 32);
                        if (od > md) { md = od; ma = oa; }
                    }
                    cm[v] = md; am[v] = ma;
                }
                unsigned clr = 0;
                if (lo) clr |= (1u << jlo);
                if (hi) clr |= (1u << jhi);
                mask &= ~clr;
            }
        }
    }
    __syncthreads();

    // ---- Merge 4 wave-lists (128 candidates) per center; wave w owns centers 4w..4w+3.
    for (int cc = 0; cc < 4; ++cc) {
        const int c = w * 4 + cc;
        float vd[WAVES]; int vi[WAVES];
#pragma unroll
        for (int r = 0; r < WAVES; ++r) { vd[r] = lds_d[r][c][lane]; vi[r] = lds_i[r][c][lane]; }

        int myBest = 0;
        for (int k = 0; k < M_; ++k) {
            float bd = vd[0]; int bi = vi[0];
#pragma unroll
            for (int r = 1; r < WAVES; ++r)
                if (vd[r] < bd || (vd[r] == bd && vi[r] < bi)) { bd = vd[r]; bi = vi[r]; }
#pragma unroll
            for (int off = 1; off < 32; off <<= 1) {   // lexicographic wave-min
                float od = __shfl_xor(bd, off, 32);
                int   oi = __shfl_xor(bi, off, 32);
                if (od < bd || (od == bd && oi < bi)) { bd = od; bi = oi; }
            }
            if (lane == k) myBest = bi;                // k-th nearest -> lane k
#pragma unroll
            for (int r = 0; r < WAVES; ++r)
                if (vd[r] == bd && vi[r] == bi) { vd[r] = INFV; vi[r] = 0x7FFFFFFF; }
        }

        const int g = gbase + c;
        const float* cptr = center + ((size_t)b * G_ + g) * 3;
        const float* pptr = xyz + ((size_t)b * N_ + myBest) * 3;
        float* op = out + (((size_t)b * G_ + g) * M_ + lane) * 3;
        op[0] = pptr[0] - cptr[0];
        op[1] = pptr[1] - cptr[1];
        op[2] = pptr[2] - cptr[2];
    }
}

extern "C" void kernel_launch(void* const* d_in, const int* in_sizes, int n_in,
                              void* d_out, int out_size, void* d_ws, size_t ws_size,
                              hipStream_t stream) {
    const float* xyz    = (const float*)d_in[0];   // B N 3
    const float* center = (const float*)d_in[1];   // B G 3
    float* out = (float*)d_out;                    // B G M 3
    dim3 grid(B_ * (G_ / 16));                     // 512 blocks
    dim3 block(128);                               // 4 waves; each wave = 16 centers x N/4 points
    knn_group_wmma_kernel<<<grid, block, 0, stream>>>(xyz, center, out);
}